// SGC_87084756893759
// MI455X (gfx1250) — compile-verified
//
#include <hip/hip_runtime.h>
#include <hip/hip_bf16.h>

typedef float v2f __attribute__((ext_vector_type(2)));
typedef float v8f __attribute__((ext_vector_type(8)));

#define NNODES 50000
#define NEDGES 600000
#define DIN    500
#define DHID   128
#define DOUT   40

// ---------------- degree / symmetric norm ----------------
__global__ void k_init_deg(float* deg, int n) {
    int i = blockIdx.x * blockDim.x + threadIdx.x;
    if (i < n) deg[i] = 1.0f;              // self loop contributes 1
}

__global__ void k_deg_edges(const long long* __restrict__ col, float* deg, int e) {
    int i = blockIdx.x * blockDim.x + threadIdx.x;
    if (i < e) atomicAdd(&deg[(int)col[i]], 1.0f);
}

__global__ void k_dinv(const float* __restrict__ deg, float* dinv, int n) {
    int i = blockIdx.x * blockDim.x + threadIdx.x;
    if (i < n) dinv[i] = rsqrtf(deg[i]);   // deg >= 1 always (self loops)
}

// ---------------- WMMA f32 GEMM: C[M x Nreal] = A[M x K] @ B[K x Nreal] ----------------
// 8 waves per block; each wave computes a 16-row strip x (NT*16) columns.
// Weight chunk staged K-contiguous (transposed) in LDS so each B fragment
// {B[kr][n], B[kr+1][n]} is one aligned ds_load_b64 straight into the WMMA
// source register pair. Row stride padded to 66 floats to spread banks.
template <int NT>
__global__ __launch_bounds__(256) void k_gemm_wmma(const float* __restrict__ A,
                                                   const float* __restrict__ B,
                                                   float* __restrict__ C,
                                                   int M, int K, int Nreal) {
    constexpr int KC   = 64;
    constexpr int KCP  = KC + 2;       // pad: lane stride 66 floats -> distinct banks
    constexpr int NPAD = NT * 16;
    __shared__ float sBt[NPAD][KCP];   // transposed: [n][k]

    const int tid   = threadIdx.x;
    const int wave  = tid >> 5;
    const int lane  = tid & 31;
    const int lhalf = lane >> 4;       // 0: lanes 0-15, 1: lanes 16-31
    const int lmod  = lane & 15;
    const int strip = blockIdx.x * 8 + wave;
    const bool active = (strip * 16) < M;      // wave-uniform
    const int m0 = strip * 16;

    v8f acc[NT] = {};

    for (int kc = 0; kc < K; kc += KC) {
        const int kclen = min(KC, K - kc);     // always a multiple of 4 here
        // cooperative stage of B[kc:kc+kclen, 0:NPAD]^T into LDS (zero-pad n >= Nreal)
        for (int idx = tid; idx < kclen * NPAD; idx += 256) {
            int k = idx / NPAD;                // global reads coalesced over n
            int n = idx % NPAD;
            sBt[n][k] = (n < Nreal) ? B[(long)(kc + k) * Nreal + n] : 0.0f;
        }
        __syncthreads();

        if (active) {
            const float* arow = A + (long)(m0 + lmod) * K + kc;
            for (int kk = 0; kk < kclen; kk += 4) {
                const int kr = kk + 2 * lhalf;          // even -> 8B aligned
                v2f a = *(const v2f*)(arow + kr);       // global_load_b64
#pragma unroll
                for (int t = 0; t < NT; ++t) {
                    v2f b = *(const v2f*)&sBt[t * 16 + lmod][kr];   // ds_load_b64
                    acc[t] = __builtin_amdgcn_wmma_f32_16x16x4_f32(
                        /*neg_a=*/false, a, /*neg_b=*/false, b,
                        /*c_mod=*/(short)0, acc[t],
                        /*reuse_a=*/false, /*reuse_b=*/false);
                }
            }
        }
        __syncthreads();
    }

    if (active) {
#pragma unroll
        for (int t = 0; t < NT; ++t) {
#pragma unroll
            for (int r = 0; r < 8; ++r) {
                int rowi = m0 + r + 8 * lhalf;
                int coln = t * 16 + lmod;
                if (coln < Nreal)
                    C[(long)rowi * Nreal + coln] = acc[t][r];
            }
        }
    }
}

// ---------------- aggregation ----------------
// out[i,d] = bias[d] + h[i,d] * dinv[i]^2   (self loop term + bias)
__global__ void k_agg_init(const float* __restrict__ h, const float* __restrict__ dinv,
                           const float* __restrict__ bias, float* __restrict__ out,
                           int n, int dim) {
    long i = (long)blockIdx.x * blockDim.x + threadIdx.x;
    long total = (long)n * dim;
    if (i < total) {
        int node = (int)(i / dim);
        int d    = (int)(i % dim);
        float dv = dinv[node];
        out[i] = bias[d] + h[i] * dv * dv;
    }
}

// out[col,d] += h[row,d] * dinv[row]*dinv[col]; 32 lanes per edge
__global__ void k_agg_edges(const float* __restrict__ h, const long long* __restrict__ row,
                            const long long* __restrict__ col, const float* __restrict__ dinv,
                            float* __restrict__ out, int e, int dim) {
    long gid  = (long)blockIdx.x * blockDim.x + threadIdx.x;
    int edge  = (int)(gid >> 5);
    int lane  = (int)(gid & 31);
    if (edge >= e) return;
    int r = (int)row[edge];
    int c = (int)col[edge];
    float nrm = dinv[r] * dinv[c];
    const float* hs = h + (long)r * dim;
    float* od = out + (long)c * dim;
    for (int d = lane; d < dim; d += 32)
        atomicAdd(&od[d], hs[d] * nrm);
}

// ---------------- log softmax (one wave32 per node) ----------------
__global__ void k_logsoftmax(const float* __restrict__ h, float* __restrict__ out,
                             int n, int dim) {
    int wave = threadIdx.x >> 5;
    int lane = threadIdx.x & 31;
    int node = blockIdx.x * 8 + wave;
    if (node >= n) return;
    const float* hr = h + (long)node * dim;
    float m = -3.4e38f;
    for (int d = lane; d < dim; d += 32) m = fmaxf(m, hr[d]);
    for (int off = 16; off; off >>= 1) m = fmaxf(m, __shfl_xor(m, off, 32));
    float s = 0.0f;
    for (int d = lane; d < dim; d += 32) s += expf(hr[d] - m);
    for (int off = 16; off; off >>= 1) s += __shfl_xor(s, off, 32);
    float lse = m + logf(s);
    for (int d = lane; d < dim; d += 32) out[(long)node * dim + d] = hr[d] - lse;
}

// ---------------- launch ----------------
extern "C" void kernel_launch(void* const* d_in, const int* in_sizes, int n_in,
                              void* d_out, int out_size, void* d_ws, size_t ws_size,
                              hipStream_t stream) {
    const float*     x  = (const float*)d_in[0];
    const long long* ei = (const long long*)d_in[1];   // int64 edge_index [2, E]
    const float*     W1 = (const float*)d_in[2];
    const float*     b1 = (const float*)d_in[3];
    const float*     W2 = (const float*)d_in[4];
    const float*     b2 = (const float*)d_in[5];
    const float*     W3 = (const float*)d_in[6];
    const float*     b3 = (const float*)d_in[7];
    float* out = (float*)d_out;

    const long long* row = ei;            // edge_index[0]
    const long long* col = ei + NEDGES;   // edge_index[1]

    float* ws   = (float*)d_ws;
    float* deg  = ws;  ws += NNODES;
    float* dinv = ws;  ws += NNODES;
    float* bufA = ws;  ws += (size_t)NNODES * DHID;   // 25.6 MB
    float* bufB = ws;  ws += (size_t)NNODES * DHID;   // 25.6 MB
    float* bufC = ws;  ws += (size_t)NNODES * DOUT;   // 8 MB
    float* bufD = ws;  ws += (size_t)NNODES * DOUT;   // 8 MB

    const int TB = 256;
    const int strips      = NNODES / 16;          // 3125 (exact)
    const int gemm_blocks = (strips + 7) / 8;     // 391
    const int edge_blocks = (int)(((long)NEDGES * 32 + TB - 1) / TB);

    // symmetric normalization
    k_init_deg <<<(NNODES + TB - 1) / TB, TB, 0, stream>>>(deg, NNODES);
    k_deg_edges<<<(NEDGES + TB - 1) / TB, TB, 0, stream>>>(col, deg, NEDGES);
    k_dinv     <<<(NNODES + TB - 1) / TB, TB, 0, stream>>>(deg, dinv, NNODES);

    // layer 1: h1 = agg(x @ W1) + b1
    k_gemm_wmma<8><<<gemm_blocks, TB, 0, stream>>>(x, W1, bufA, NNODES, DIN, DHID);
    k_agg_init <<<(int)(((long)NNODES * DHID + TB - 1) / TB), TB, 0, stream>>>(bufA, dinv, b1, bufB, NNODES, DHID);
    k_agg_edges<<<edge_blocks, TB, 0, stream>>>(bufA, row, col, dinv, bufB, NEDGES, DHID);

    // layer 2
    k_gemm_wmma<8><<<gemm_blocks, TB, 0, stream>>>(bufB, W2, bufA, NNODES, DHID, DHID);
    k_agg_init <<<(int)(((long)NNODES * DHID + TB - 1) / TB), TB, 0, stream>>>(bufA, dinv, b2, bufB, NNODES, DHID);
    k_agg_edges<<<edge_blocks, TB, 0, stream>>>(bufA, row, col, dinv, bufB, NEDGES, DHID);

    // layer 3 (N=40, NT=3 with column guards)
    k_gemm_wmma<3><<<gemm_blocks, TB, 0, stream>>>(bufB, W3, bufC, NNODES, DHID, DOUT);
    k_agg_init <<<(int)(((long)NNODES * DOUT + TB - 1) / TB), TB, 0, stream>>>(bufC, dinv, b3, bufD, NNODES, DOUT);
    k_agg_edges<<<edge_blocks, TB, 0, stream>>>(bufC, row, col, dinv, bufD, NEDGES, DOUT);

    // log softmax
    k_logsoftmax<<<(NNODES + 7) / 8, TB, 0, stream>>>(bufD, out, NNODES, DOUT);
}